// DeeperGCN_83219286328197
// MI455X (gfx1250) — compile-verified
//
#include <hip/hip_runtime.h>
#include <cstdint>

#define HC 64
#define HC2 128
#define NLAYERS 4

typedef float v2f __attribute__((ext_vector_type(2)));
typedef float v8f __attribute__((ext_vector_type(8)));

__device__ __forceinline__ float warp_sum32(float v) {
#pragma unroll
    for (int m = 16; m > 0; m >>= 1) v += __shfl_xor(v, m, 32);
    return v;
}

// Order-preserving float -> uint key so segment-max lowers to global_atomic_max_u32.
__device__ __forceinline__ unsigned fkey(float f) {
    unsigned u = __float_as_uint(f);
    return (u & 0x80000000u) ? ~u : (u | 0x80000000u);
}
__device__ __forceinline__ float funkey(unsigned k) {
    unsigned u = (k & 0x80000000u) ? (k & 0x7fffffffu) : ~k;
    return __uint_as_float(u);
}

// ---------------- h0 = x @ node_W + node_b ----------------
__global__ void k_embed(const float* __restrict__ x, const float* __restrict__ nW,
                        const float* __restrict__ nb, float* __restrict__ h, int N) {
    int idx = blockIdx.x * blockDim.x + threadIdx.x;
    if (idx >= N * HC) return;
    int n = idx >> 6, c = idx & 63;
    float v = x[n * 3 + 0] * nW[0 * HC + c] + x[n * 3 + 1] * nW[1 * HC + c] +
              x[n * 3 + 2] * nW[2 * HC + c] + nb[c];
    h[idx] = v;
}

// ------- z = (layer? relu(LN(h)) : h); reset max/denom/wsum accumulators -------
__global__ void k_prep(const float* __restrict__ h, const float* __restrict__ g,
                       const float* __restrict__ b, float* __restrict__ z,
                       unsigned* __restrict__ mkey, float* __restrict__ denom,
                       float* __restrict__ wsum, int N, int doNorm) {
    int wave = threadIdx.x >> 5, lane = threadIdx.x & 31;
    int node = blockIdx.x * 8 + wave;
    if (node >= N) return;
    int o = node * HC, c0 = lane, c1 = lane + 32;
    float v0 = h[o + c0], v1 = h[o + c1], z0, z1;
    if (doNorm) {
        float mean = warp_sum32(v0 + v1) * (1.0f / 64.0f);
        float d0 = v0 - mean, d1 = v1 - mean;
        float var = warp_sum32(d0 * d0 + d1 * d1) * (1.0f / 64.0f);
        float r = rsqrtf(var + 1e-5f);
        z0 = fmaxf(d0 * r * g[c0] + b[c0], 0.0f);
        z1 = fmaxf(d1 * r * g[c1] + b[c1], 0.0f);
    } else { z0 = v0; z1 = v1; }
    z[o + c0] = z0; z[o + c1] = z1;
    mkey[o + c0] = 0u;  mkey[o + c1] = 0u;
    denom[o + c0] = 0.f; denom[o + c1] = 0.f;
    wsum[o + c0] = 0.f;  wsum[o + c1] = 0.f;
}

// ---------------- edge pass A: per-(dst,c) max of s ----------------
__global__ void k_edge_max(const long long* __restrict__ ei, const float* __restrict__ eattr,
                           const float* __restrict__ eW, const float* __restrict__ eb,
                           const float* __restrict__ z, const float* __restrict__ t, int layer,
                           unsigned* __restrict__ mkey, int E) {
    long long gid = (long long)blockIdx.x * blockDim.x + threadIdx.x;
    int e = (int)(gid >> 5);
    if (e >= E) return;
    int lane = threadIdx.x & 31;
    int src = (int)ei[e];
    int dst = (int)ei[(size_t)E + (size_t)e];
    float w = eattr[e], tv = t[layer];
    int c0 = lane, c1 = lane + 32;
    float m0 = fmaxf(z[src * HC + c0] + w * eW[c0] + eb[c0], 0.0f) + 1e-7f;
    float m1 = fmaxf(z[src * HC + c1] + w * eW[c1] + eb[c1], 0.0f) + 1e-7f;
    atomicMax(&mkey[dst * HC + c0], fkey(m0 * tv));
    atomicMax(&mkey[dst * HC + c1], fkey(m1 * tv));
}

// ---------------- edge pass B: exp(s-m) accumulation ----------------
__global__ void k_edge_accum(const long long* __restrict__ ei, const float* __restrict__ eattr,
                             const float* __restrict__ eW, const float* __restrict__ eb,
                             const float* __restrict__ z, const float* __restrict__ t, int layer,
                             const unsigned* __restrict__ mkey, float* __restrict__ denom,
                             float* __restrict__ wsum, int E) {
    long long gid = (long long)blockIdx.x * blockDim.x + threadIdx.x;
    int e = (int)(gid >> 5);
    if (e >= E) return;
    int lane = threadIdx.x & 31;
    int src = (int)ei[e];
    int dst = (int)ei[(size_t)E + (size_t)e];
    float w = eattr[e], tv = t[layer];
#pragma unroll
    for (int half = 0; half < 2; ++half) {
        int c = lane + half * 32;
        float msg = fmaxf(z[src * HC + c] + w * eW[c] + eb[c], 0.0f) + 1e-7f;
        float s = msg * tv;
        float m = funkey(mkey[dst * HC + c]);
        if (!(m > -3.0e38f && m < 3.0e38f)) m = 0.0f;  // no in-edges -> NaN key -> 0
        float a = __expf(s - m);
        atomicAdd(&denom[dst * HC + c], a);
        atomicAdd(&wsum[dst * HC + c], a * msg);
    }
}

// ------- fused: out = agg+z; h1 = relu(LN(out@W1+b1)); h += h1@W2+b2 (f32 WMMA) -------
__global__ void k_node_update(const float* __restrict__ z, const float* __restrict__ denom,
                              const float* __restrict__ wsum, float* __restrict__ h,
                              const float* __restrict__ W1, const float* __restrict__ b1,
                              const float* __restrict__ g1, const float* __restrict__ bt1,
                              const float* __restrict__ W2, const float* __restrict__ b2,
                              int N, int firstLayer) {
    __shared__ float sOut[64][65];
    __shared__ float sH1[64][129];
    int tid = threadIdx.x;
    int base = blockIdx.x * 64;

    // Phase 1: build 64x64 "out" tile (agg + residual), zero-fill past N
#pragma unroll
    for (int i = 0; i < 16; ++i) {
        int idx = tid + i * 256;
        int m = idx >> 6, c = idx & 63;
        int node = base + m;
        float val = 0.0f;
        if (node < N) {
            int o = node * HC + c;
            val = wsum[o] / fmaxf(denom[o], 1e-16f) + z[o];
        }
        sOut[m][c] = val;
    }
    __syncthreads();

    int wave = tid >> 5, lane = tid & 31, ln = lane & 15;
    int kh = (lane >> 4) * 2;        // K offset within a 16x4 f32 A/B fragment
    int rsel = (lane >> 4) << 3;     // +8 row offset for high half-wave in C/D

    // GEMM1: sH1 = sOut(64x64) @ W1(64x128) + b1  — wave w owns column tile w
    {
        int n = wave * 16 + ln;
        float bn = b1[n];
        for (int rt = 0; rt < 4; ++rt) {
            v8f acc = {0.f, 0.f, 0.f, 0.f, 0.f, 0.f, 0.f, 0.f};
            int m = rt * 16 + ln;
#pragma unroll
            for (int k0 = 0; k0 < 64; k0 += 4) {
                v2f a, bfrag;
                a.x = sOut[m][k0 + kh];
                a.y = sOut[m][k0 + kh + 1];
                bfrag.x = W1[(k0 + kh) * HC2 + n];
                bfrag.y = W1[(k0 + kh + 1) * HC2 + n];
                acc = __builtin_amdgcn_wmma_f32_16x16x4_f32(
                    false, a, false, bfrag, (short)0, acc, false, false);
            }
#pragma unroll
            for (int r = 0; r < 8; ++r) sH1[rt * 16 + r + rsel][n] = acc[r] + bn;
        }
    }
    __syncthreads();

    // LayerNorm(128) + ReLU per row; wave handles 8 rows, lane 4 channels/row
    for (int j = 0; j < 8; ++j) {
        int row = wave * 8 + j;
        float v[4], s = 0.0f;
#pragma unroll
        for (int q = 0; q < 4; ++q) { v[q] = sH1[row][lane + 32 * q]; s += v[q]; }
        float mean = warp_sum32(s) * (1.0f / 128.0f);
        float vs = 0.0f;
#pragma unroll
        for (int q = 0; q < 4; ++q) { v[q] -= mean; vs += v[q] * v[q]; }
        float var = warp_sum32(vs) * (1.0f / 128.0f);
        float r = rsqrtf(var + 1e-5f);
#pragma unroll
        for (int q = 0; q < 4; ++q) {
            int c = lane + 32 * q;
            sH1[row][c] = fmaxf(v[q] * r * g1[c] + bt1[c], 0.0f);
        }
    }
    __syncthreads();

    // GEMM2: h2 = sH1(64x128) @ W2(128x64) + b2; h = firstLayer ? h2 : h + h2
    for (int tp = 0; tp < 2; ++tp) {
        int tile = wave + tp * 8;
        int rt = tile >> 2, ct = tile & 3;
        int n = ct * 16 + ln;
        int m = rt * 16 + ln;
        v8f acc = {0.f, 0.f, 0.f, 0.f, 0.f, 0.f, 0.f, 0.f};
#pragma unroll
        for (int k0 = 0; k0 < 128; k0 += 4) {
            v2f a, bfrag;
            a.x = sH1[m][k0 + kh];
            a.y = sH1[m][k0 + kh + 1];
            bfrag.x = W2[(k0 + kh) * HC + n];
            bfrag.y = W2[(k0 + kh + 1) * HC + n];
            acc = __builtin_amdgcn_wmma_f32_16x16x4_f32(
                false, a, false, bfrag, (short)0, acc, false, false);
        }
        float bn = b2[n];
#pragma unroll
        for (int r = 0; r < 8; ++r) {
            int mr = rt * 16 + r + rsel;
            int node = base + mr;
            if (node < N) {
                int o = node * HC + n;
                float val = acc[r] + bn;
                h[o] = firstLayer ? val : (h[o] + val);
            }
        }
    }
}

// ---------------- out = relu(LN(h, g0, b0)) @ lin_W + lin_b ----------------
__global__ void k_final(const float* __restrict__ h, const float* __restrict__ g,
                        const float* __restrict__ b, const float* __restrict__ lw,
                        const float* __restrict__ lb, float* __restrict__ out, int N) {
    int wave = threadIdx.x >> 5, lane = threadIdx.x & 31;
    int node = blockIdx.x * 8 + wave;
    if (node >= N) return;
    int o = node * HC, c0 = lane, c1 = lane + 32;
    float v0 = h[o + c0], v1 = h[o + c1];
    float mean = warp_sum32(v0 + v1) * (1.0f / 64.0f);
    float d0 = v0 - mean, d1 = v1 - mean;
    float var = warp_sum32(d0 * d0 + d1 * d1) * (1.0f / 64.0f);
    float r = rsqrtf(var + 1e-5f);
    float z0 = fmaxf(d0 * r * g[c0] + b[c0], 0.0f);
    float z1 = fmaxf(d1 * r * g[c1] + b[c1], 0.0f);
    float tot = warp_sum32(z0 * lw[c0] + z1 * lw[c1]);
    if (lane == 0) out[node] = tot + lb[0];
}

extern "C" void kernel_launch(void* const* d_in, const int* in_sizes, int n_in,
                              void* d_out, int out_size, void* d_ws, size_t ws_size,
                              hipStream_t stream) {
    const float*     x      = (const float*)d_in[0];
    const long long* ei     = (const long long*)d_in[1];
    const float*     eattr  = (const float*)d_in[2];
    const float*     node_W = (const float*)d_in[3];
    const float*     node_b = (const float*)d_in[4];
    const float*     edge_W = (const float*)d_in[5];
    const float*     edge_b = (const float*)d_in[6];
    const float*     t      = (const float*)d_in[7];
    const float*     cW1    = (const float*)d_in[8];
    const float*     cb1    = (const float*)d_in[9];
    const float*     cg1    = (const float*)d_in[10];
    const float*     cbt1   = (const float*)d_in[11];
    const float*     cW2    = (const float*)d_in[12];
    const float*     cb2    = (const float*)d_in[13];
    const float*     norm_g = (const float*)d_in[14];
    const float*     norm_b = (const float*)d_in[15];
    const float*     lin_W  = (const float*)d_in[16];
    const float*     lin_b  = (const float*)d_in[17];
    float* out = (float*)d_out;

    const int N = in_sizes[0] / 3;
    const int E = in_sizes[2];

    // Workspace: h | z | mkey | denom | wsum  (each N*HC)
    float*    h     = (float*)d_ws;
    float*    z     = h + (size_t)N * HC;
    unsigned* mkey  = (unsigned*)(z + (size_t)N * HC);
    float*    denom = (float*)(mkey + (size_t)N * HC);
    float*    wsum  = denom + (size_t)N * HC;

    const int nodeBlocks  = (N + 7) / 8;
    const int embedBlocks = (N * HC + 255) / 256;
    const int edgeBlocks  = (E + 7) / 8;          // 1 wave per edge, 8 waves/block
    const int tileBlocks  = (N + 63) / 64;

    k_embed<<<embedBlocks, 256, 0, stream>>>(x, node_W, node_b, h, N);

    for (int l = 0; l < NLAYERS; ++l) {
        k_prep<<<nodeBlocks, 256, 0, stream>>>(
            h, norm_g + l * HC, norm_b + l * HC, z, mkey, denom, wsum, N, l > 0 ? 1 : 0);
        k_edge_max<<<edgeBlocks, 256, 0, stream>>>(
            ei, eattr, edge_W, edge_b, z, t, l, mkey, E);
        k_edge_accum<<<edgeBlocks, 256, 0, stream>>>(
            ei, eattr, edge_W, edge_b, z, t, l, mkey, denom, wsum, E);
        k_node_update<<<tileBlocks, 256, 0, stream>>>(
            z, denom, wsum, h,
            cW1 + (size_t)l * HC * HC2, cb1 + l * HC2, cg1 + l * HC2, cbt1 + l * HC2,
            cW2 + (size_t)l * HC2 * HC, cb2 + l * HC, N, l == 0 ? 1 : 0);
    }

    k_final<<<nodeBlocks, 256, 0, stream>>>(h, norm_g, norm_b, lin_W, lin_b, out, N);
}